// MultiHeadSelfAttention_80616536146599
// MI455X (gfx1250) — compile-verified
//
#include <hip/hip_runtime.h>

// Multi-head self-attention score matrix, mean over heads.
// N=4096, IN_F=256, H=8, D=64, logit scale = 1/(8*0.5) = 0.25.
// All matrix math via V_WMMA_F32_16X16X4_F32 (true fp32 WMMA on CDNA5).
// B operand staged global->LDS with GLOBAL_LOAD_ASYNC_TO_LDS_B128 (ASYNCcnt),
// double-buffered so the copy overlaps the WMMA stream.
//
// Workspace:
//   Q    [4096][512]    row-major            8 MB
//   KtP  [256][4096][2] pair-interleaved Kt  8 MB  (KtP[d>>1][m][d&1] = K[m][d])
//   Lsum [4096][8]      softmax denominators 128 KB

#define N_NODES 4096
#define IN_F    256
#define HD      512
#define NHEADS  8
#define DHEAD   64
#define INV_SCALE 0.25f

typedef float v2f __attribute__((ext_vector_type(2)));
typedef float v8f __attribute__((ext_vector_type(8)));
typedef int   vint4 __attribute__((vector_size(16)));   // matches builtin's V4i

#if defined(__has_builtin)
#if __has_builtin(__builtin_amdgcn_global_load_async_to_lds_b128) && \
    __has_builtin(__builtin_amdgcn_s_wait_asynccnt)
#define HAVE_ASYNC_LDS 1
#endif
#endif
#ifndef HAVE_ASYNC_LDS
#define HAVE_ASYNC_LDS 0
#endif

static __device__ __forceinline__ v8f wmma_f32(v2f a, v2f b, v8f c) {
    // (neg_a, A, neg_b, B, c_mod, C, reuse_a, reuse_b)
    return __builtin_amdgcn_wmma_f32_16x16x4_f32(false, a, false, b,
                                                 (short)0, c, false, false);
}

// Copy 16 bytes global -> LDS. Async (ASYNCcnt) when available.
static __device__ __forceinline__ void cp16_g2l(const float* src, float* ldsDst) {
#if HAVE_ASYNC_LDS
    __builtin_amdgcn_global_load_async_to_lds_b128(
        (__attribute__((address_space(1))) vint4*)(uintptr_t)src,
        (__attribute__((address_space(3))) vint4*)(uintptr_t)ldsDst, 0, 0);
#else
    *(float4*)ldsDst = *(const float4*)src;
#endif
}

template <int CNT>
static __device__ __forceinline__ void async_wait() {
#if HAVE_ASYNC_LDS
    __builtin_amdgcn_s_wait_asynccnt(CNT);
#endif
}

// ---------------------------------------------------------------------------
// Projection GEMMs: X[4096,256] @ W[256,512] + bias.  One wave per 16x16 tile.
// ---------------------------------------------------------------------------
__global__ __launch_bounds__(256)
void proj_q_kernel(const float* __restrict__ X, const float* __restrict__ W,
                   const float* __restrict__ bias, float* __restrict__ Q)
{
    const int lane = threadIdx.x & 31;
    const int wave = threadIdx.x >> 5;
    const int tile = blockIdx.x * 8 + wave;   // 8192 tiles = 256 x 32
    const int tr = tile >> 5, tc = tile & 31;
    const int half = lane >> 4, l16 = lane & 15;
    const int arow = tr * 16 + l16;
    const int bcol = tc * 16 + l16;

    v8f c = {};
#pragma unroll 8
    for (int k = 0; k < IN_F; k += 4) {
        const int ka = k + 2 * half;
        v2f a = *(const v2f*)&X[arow * IN_F + ka];
        v2f b;
        b.x = W[ka * HD + bcol];
        b.y = W[(ka + 1) * HD + bcol];
        c = wmma_f32(a, b, c);
    }
    const float bv = bias[bcol];
#pragma unroll
    for (int r = 0; r < 8; ++r) {
        const int row = tr * 16 + r + 8 * half;
        Q[(size_t)row * HD + bcol] = c[r] + bv;
    }
}

__global__ __launch_bounds__(256)
void proj_kt_kernel(const float* __restrict__ X, const float* __restrict__ W,
                    const float* __restrict__ bias, float* __restrict__ KtP)
{
    const int lane = threadIdx.x & 31;
    const int wave = threadIdx.x >> 5;
    const int tile = blockIdx.x * 8 + wave;
    const int tr = tile >> 5, tc = tile & 31;   // tr: node tile, tc: hd tile
    const int half = lane >> 4, l16 = lane & 15;
    const int arow = tr * 16 + l16;
    const int bcol = tc * 16 + l16;             // hd index d

    v8f c = {};
#pragma unroll 8
    for (int k = 0; k < IN_F; k += 4) {
        const int ka = k + 2 * half;
        v2f a = *(const v2f*)&X[arow * IN_F + ka];
        v2f b;
        b.x = W[ka * HD + bcol];
        b.y = W[(ka + 1) * HD + bcol];
        c = wmma_f32(a, b, c);
    }
    const float bv = bias[bcol];
    const size_t base = (size_t)(bcol >> 1) * (2 * N_NODES) + (bcol & 1);
#pragma unroll
    for (int r = 0; r < 8; ++r) {
        const int m = tr * 16 + r + 8 * half;
        KtP[base + 2 * m] = c[r] + bv;          // KtP[d>>1][m][d&1]
    }
}

// ---------------------------------------------------------------------------
// Kernel 2: L[n][h] = sum_m exp(0.25 * Q[n,h,:].K[m,h,:])
// Block = 8 waves, one head; wave w owns n-tile. Kt strip (64d x 64m, paired)
// async-staged to double-buffered LDS; per-lane partial sums, reduce once.
// ---------------------------------------------------------------------------
__global__ __launch_bounds__(256)
void denom_kernel(const float* __restrict__ Q, const float* __restrict__ KtP,
                  float* __restrict__ Lsum)
{
    __shared__ float ldsB[2][32 * 128];         // 2 x 16 KB

    const int lane = threadIdx.x & 31;
    const int wave = threadIdx.x >> 5;
    const int h    = blockIdx.x & 7;
    const int nt   = (blockIdx.x >> 3) * 8 + wave;   // 0..255
    const int half = lane >> 4, l16 = lane & 15;
    const int arow = nt * 16 + l16;
    const int p0   = h * 32;

    // Preload 16 A fragments (Q strip for this n-tile/head), reused 256x.
    v2f aF[16];
#pragma unroll
    for (int s = 0; s < 16; ++s) {
        const int kk = h * DHEAD + s * 4 + 2 * half;
        aF[s] = *(const v2f*)&Q[arow * HD + kk];
    }

    float lrun[8];
#pragma unroll
    for (int r = 0; r < 8; ++r) lrun[r] = 0.f;

    const int srow = threadIdx.x >> 3;          // 0..31 pair-row
    const int schk = threadIdx.x & 7;           // 0..7  16-float chunk
    const float* gbase = &KtP[(size_t)(p0 + srow) * (2 * N_NODES) + schk * 16];
    float* const dst0 = &ldsB[0][srow * 128 + schk * 16];
    float* const dst1 = &ldsB[1][srow * 128 + schk * 16];

    {   // prologue: stage group 0 (4 x b128 per thread = 16 KB per block)
#pragma unroll
        for (int q = 0; q < 4; ++q) cp16_g2l(gbase + 4 * q, dst0 + 4 * q);
    }

    for (int g = 0; g < N_NODES / 64; ++g) {
        const int cur = g & 1;
        if (g + 1 < N_NODES / 64) {
            const float* s = gbase + (size_t)(g + 1) * 128;
            float* d = cur ? dst0 : dst1;
#pragma unroll
            for (int q = 0; q < 4; ++q) cp16_g2l(s + 4 * q, d + 4 * q);
            async_wait<4>();                    // group g landed (in-order)
        } else {
            async_wait<0>();
        }
        __syncthreads();                        // group g visible to all waves

        const float* buf = ldsB[cur];
#pragma unroll
        for (int j = 0; j < 4; ++j) {           // 4 m-tiles of 16
            const int colL = j * 16 + l16;
            v8f c = {};
#pragma unroll
            for (int s = 0; s < 16; ++s) {
                const int pairL = s * 2 + half;
                v2f b = *(const v2f*)&buf[pairL * 128 + colL * 2];
                c = wmma_f32(aF[s], b, c);
            }
#pragma unroll
            for (int r = 0; r < 8; ++r)         // per-lane partial sum only
                lrun[r] += __expf(c[r] * INV_SCALE);
        }
        __syncthreads();                        // done reading buf before reuse
    }

    // One cross-lane reduction at the end (sum over the 16 lanes per C row).
#pragma unroll
    for (int r = 0; r < 8; ++r)
        for (int o = 1; o < 16; o <<= 1)
            lrun[r] += __shfl_xor(lrun[r], o, 32);

    if (l16 == 0) {
#pragma unroll
        for (int r = 0; r < 8; ++r) {
            const int row = nt * 16 + r + 8 * half;
            Lsum[row * NHEADS + h] = lrun[r];
        }
    }
}

// ---------------------------------------------------------------------------
// Kernel 3: out[n][m] = (1/8) * sum_h exp(0.25*s_nhm) / L[n][h]
// Block = 8 waves: 128 rows x 64 cols, loop over heads; Kt strip per head
// async-staged to double-buffered LDS.
// ---------------------------------------------------------------------------
__global__ __launch_bounds__(256)
void out_kernel(const float* __restrict__ Q, const float* __restrict__ KtP,
                const float* __restrict__ Lsum, float* __restrict__ out)
{
    __shared__ float ldsB[2][32 * 128];         // 2 x 16 KB

    const int lane = threadIdx.x & 31;
    const int wave = threadIdx.x >> 5;
    const int mg   = blockIdx.x & 63;           // m-group: 64 cols
    const int nt   = (blockIdx.x >> 6) * 8 + wave;
    const int half = lane >> 4, l16 = lane & 15;
    const int arow = nt * 16 + l16;
    const int c0   = mg * 64;

    float acc[4][8];
#pragma unroll
    for (int j = 0; j < 4; ++j)
#pragma unroll
        for (int r = 0; r < 8; ++r) acc[j][r] = 0.f;

    const int srow = threadIdx.x >> 3;
    const int schk = threadIdx.x & 7;
    const float* gbase = &KtP[(size_t)srow * (2 * N_NODES) + c0 * 2 + schk * 16];
    float* const dst0 = &ldsB[0][srow * 128 + schk * 16];
    float* const dst1 = &ldsB[1][srow * 128 + schk * 16];
    const size_t hstride = (size_t)32 * 2 * N_NODES;    // pair-rows per head

    {   // prologue: stage head 0
#pragma unroll
        for (int q = 0; q < 4; ++q) cp16_g2l(gbase + 4 * q, dst0 + 4 * q);
    }

    for (int h = 0; h < NHEADS; ++h) {
        const int cur = h & 1;
        if (h + 1 < NHEADS) {
            const float* s = gbase + (size_t)(h + 1) * hstride;
            float* d = cur ? dst0 : dst1;
#pragma unroll
            for (int q = 0; q < 4; ++q) cp16_g2l(s + 4 * q, d + 4 * q);
            async_wait<4>();
        } else {
            async_wait<0>();
        }
        __syncthreads();

        // A fragments for this head.
        v2f aF[16];
#pragma unroll
        for (int s = 0; s < 16; ++s) {
            const int kk = h * DHEAD + s * 4 + 2 * half;
            aF[s] = *(const v2f*)&Q[arow * HD + kk];
        }
        // Softmax denominators for my 8 rows at this head.
        float iL[8];
#pragma unroll
        for (int r = 0; r < 8; ++r)
            iL[r] = 1.0f / Lsum[(nt * 16 + r + 8 * half) * NHEADS + h];

        const float* buf = ldsB[cur];
#pragma unroll
        for (int j = 0; j < 4; ++j) {
            const int colL = j * 16 + l16;
            v8f c = {};
#pragma unroll
            for (int s = 0; s < 16; ++s) {
                const int pairL = s * 2 + half;
                v2f b = *(const v2f*)&buf[pairL * 128 + colL * 2];
                c = wmma_f32(aF[s], b, c);
            }
#pragma unroll
            for (int r = 0; r < 8; ++r)
                acc[j][r] += __expf(c[r] * INV_SCALE) * iL[r];
        }
        __syncthreads();
    }

#pragma unroll
    for (int j = 0; j < 4; ++j)
#pragma unroll
        for (int r = 0; r < 8; ++r) {
            const int row = nt * 16 + r + 8 * half;
            out[(size_t)row * N_NODES + c0 + j * 16 + l16] = acc[j][r] * 0.125f;
        }
}

// ---------------------------------------------------------------------------
extern "C" void kernel_launch(void* const* d_in, const int* in_sizes, int n_in,
                              void* d_out, int out_size, void* d_ws, size_t ws_size,
                              hipStream_t stream)
{
    const float* X  = (const float*)d_in[0];
    const float* Wq = (const float*)d_in[1];
    const float* bq = (const float*)d_in[2];
    const float* Wk = (const float*)d_in[3];
    const float* bk = (const float*)d_in[4];
    float* out = (float*)d_out;

    char* ws = (char*)d_ws;
    float* Qbuf = (float*)ws;                                      // 8 MB
    float* KtP  = (float*)(ws + (size_t)N_NODES * HD * 4);         // 8 MB
    float* Lsum = (float*)(ws + (size_t)2 * N_NODES * HD * 4);     // 128 KB

    const dim3 blk(256);
    proj_q_kernel <<<1024, blk, 0, stream>>>(X, Wq, bq, Qbuf);
    proj_kt_kernel<<<1024, blk, 0, stream>>>(X, Wk, bk, KtP);
    denom_kernel<<<256, blk, 0, stream>>>(Qbuf, KtP, Lsum);   // 32 nt-groups x 8 heads
    out_kernel<<<2048, blk, 0, stream>>>(Qbuf, KtP, Lsum, out); // 32 x 64 groups
}